// ConvCNP_Encoder_65755949301956
// MI455X (gfx1250) — compile-verified
//
#include <hip/hip_runtime.h>

typedef __attribute__((ext_vector_type(2))) float v2f;
typedef __attribute__((ext_vector_type(8))) float v8f;

#define NPTS   4096
#define NGRID  65536
#define LOG2E  1.4426950408889634f

#if __has_builtin(__builtin_amdgcn_exp2f)
#define EXP2F(x) __builtin_amdgcn_exp2f(x)
#else
#define EXP2F(x) exp2f(x)
#endif

// One wave (32 lanes) handles 16 grid rows x all 4096 data points.
// Per 16-wide j-tile: one V_WMMA_F32_16X16X4_F32 produces the 16x16 tile of
//   -0.5 * ||g - x||^2 * log2(e)
// via the augmented rank-4 factorization; v_exp_f32 turns it into the Gram
// tile; VALU FMAs accumulate density / y0 / y1 feature columns.
__global__ __launch_bounds__(256) void convcnp_encoder_kernel(
    const float* __restrict__ X,     // 4096 x 2
    const float* __restrict__ Y,     // 4096 x 2
    const float* __restrict__ G,     // 65536 x 2 (grid)
    float* __restrict__ out)         // 3 x 256 x 256
{
  const int lane = threadIdx.x & 31;
  const int wave = (blockIdx.x * blockDim.x + threadIdx.x) >> 5;
  const int row0 = wave << 4;          // first of 16 grid rows for this wave
  const int n    = lane & 15;          // row-in-tile (A) / col-in-tile (B,C)
  const bool hi  = lane >= 16;

  // ---- A operand: 16x4 f32 (rows = grid pts), cols = [gx, gy, -0.5||g||^2, 1]
  // Layout: VGPR0 = K0 (lanes 0-15) / K2 (lanes 16-31); VGPR1 = K1 / K3.
  const int gi = row0 + n;
  const float gx = G[2 * gi + 0];
  const float gy = G[2 * gi + 1];
  v2f A;
  A.x = hi ? (-0.5f * (gx * gx + gy * gy)) : gx;
  A.y = hi ? 1.0f : gy;

  float acc0[8], acc1[8], acc2[8];
#pragma unroll
  for (int r = 0; r < 8; ++r) { acc0[r] = 0.f; acc1[r] = 0.f; acc2[r] = 0.f; }

  for (int jt = 0; jt < NPTS / 16; ++jt) {
    const int j = jt * 16 + n;
    const float xx = X[2 * j + 0];
    const float xy = X[2 * j + 1];
    const float y0 = Y[2 * j + 0];
    const float y1 = Y[2 * j + 1];

    // ---- B operand: 4x16 f32, col j = log2e * [xx, xy, 1, -0.5||x||^2]
    v2f B;
    B.x = LOG2E * (hi ? 1.0f : xx);
    B.y = LOG2E * (hi ? (-0.5f * (xx * xx + xy * xy)) : xy);

    v8f C = {};
    C = __builtin_amdgcn_wmma_f32_16x16x4_f32(
        /*neg_a=*/false, A, /*neg_b=*/false, B,
        /*c_mod=*/(short)0, C, /*reuse_a=*/false, /*reuse_b=*/false);

    // C[r] = log2 of Gaussian at (M = r + 8*hi, N = n); exponentiate + accumulate.
#pragma unroll
    for (int r = 0; r < 8; ++r) {
      const float g = EXP2F(C[r]);
      acc0[r] += g;
      acc1[r] = fmaf(g, y0, acc1[r]);
      acc2[r] = fmaf(g, y1, acc2[r]);
    }
  }

  // Reduce the 16 columns: butterfly within each 16-lane half of the wave.
#pragma unroll
  for (int m = 1; m < 16; m <<= 1) {
#pragma unroll
    for (int r = 0; r < 8; ++r) {
      acc0[r] += __shfl_xor(acc0[r], m, 32);
      acc1[r] += __shfl_xor(acc1[r], m, 32);
      acc2[r] += __shfl_xor(acc2[r], m, 32);
    }
  }

  // Lane n==0 of each half writes rows M = r + 8*hi.
  if (n == 0) {
#pragma unroll
    for (int r = 0; r < 8; ++r) {
      const int i   = row0 + r + (hi ? 8 : 0);           // flat grid index
      const int idx = ((i & 255) << 8) | (i >> 8);       // out[c, iy, ix]
      const float d   = acc0[r];
      const float inv = 1.0f / d;
      out[idx]              = d;
      out[NGRID + idx]      = acc1[r] * inv;
      out[2 * NGRID + idx]  = acc2[r] * inv;
    }
  }
}

extern "C" void kernel_launch(void* const* d_in, const int* in_sizes, int n_in,
                              void* d_out, int out_size, void* d_ws, size_t ws_size,
                              hipStream_t stream) {
  const float* X = (const float*)d_in[0];   // 4096 x 2
  const float* Y = (const float*)d_in[1];   // 4096 x 2
  const float* G = (const float*)d_in[2];   // 65536 x 2
  float* out = (float*)d_out;               // 3 x 256 x 256

  // 4096 waves (one per 16 grid rows) = 512 blocks x 256 threads (8 waves).
  convcnp_encoder_kernel<<<512, 256, 0, stream>>>(X, Y, G, out);
}